// RecyclingEmbedder_19602230739890
// MI455X (gfx1250) — compile-verified
//
#include <hip/hip_runtime.h>
#include <hip/hip_bf16.h>

typedef __attribute__((ext_vector_type(2))) float v2f;
typedef __attribute__((ext_vector_type(8))) float v8f;

#define NSEQ 1024
#define DIM 128
#define NBINS 37
#define LN_EPS 1e-5f

// ---------------------------------------------------------------------------
// Kernel 1: single_out = layernorm(single)   [1024 rows x 128]
// One wave32 per row: 4 floats/lane, shfl_xor tree reduction.
// ---------------------------------------------------------------------------
__global__ __launch_bounds__(256) void ln_single_kernel(
    const float* __restrict__ single, const float* __restrict__ gamma,
    const float* __restrict__ beta, float* __restrict__ out) {
  const int wave = threadIdx.x >> 5;
  const int lane = threadIdx.x & 31;
  const int row = blockIdx.x * 8 + wave;   // grid = 128 blocks -> 1024 rows
  const int c = lane * 4;
  const size_t off = (size_t)row * DIM + c;

  float4 v = *(const float4*)(single + off);
  float s = v.x + v.y + v.z + v.w;
  float sq = v.x * v.x + v.y * v.y + v.z * v.z + v.w * v.w;
#pragma unroll
  for (int m = 16; m > 0; m >>= 1) {
    s += __shfl_xor(s, m, 32);
    sq += __shfl_xor(sq, m, 32);
  }
  const float mu = s * (1.0f / DIM);
  const float var = sq * (1.0f / DIM) - mu * mu;
  const float rstd = rsqrtf(var + LN_EPS);

  float4 g = *(const float4*)(gamma + c);
  float4 b = *(const float4*)(beta + c);
  float4 o;
  o.x = (v.x - mu) * rstd * g.x + b.x;
  o.y = (v.y - mu) * rstd * g.y + b.y;
  o.z = (v.z - mu) * rstd * g.z + b.z;
  o.w = (v.w - mu) * rstd * g.w + b.w;
  *(float4*)(out + off) = o;
}

// ---------------------------------------------------------------------------
// Kernel 2: pair_out = layernorm(pair) + distogram_emb
// Each block owns a 16x16 (i,j) tile:
//   step 1: wave 0 computes the 16x16 Gram tile x_i . x_j with ONE
//           v_wmma_f32_16x16x4_f32 (K = {x,y,z,0}), derives d, bins -> LDS.
//   step 2: 8 waves stream 256 pair rows (512B/row, float4 per lane),
//           layernorm via shfl_xor reduction, add wT[bin] + bias from LDS.
// Memory-bound streaming: 1 read + 1 write of the 512MB pair tensor.
// ---------------------------------------------------------------------------
__global__ __launch_bounds__(256) void pair_kernel(
    const float* __restrict__ x, const float* __restrict__ pair,
    const float* __restrict__ weight,      // [128][37] row-major
    const float* __restrict__ bias,        // [128]
    const float* __restrict__ gamma_pair,  // [128]
    const float* __restrict__ beta_pair,   // [128]
    float* __restrict__ out_pair) {
  __shared__ float sWT[NBINS * DIM];  // weight transposed: sWT[b*128+c]
  __shared__ float sXI[16 * 4];       // x,y,z,|x|^2 for tile rows i
  __shared__ float sXJ[16 * 4];       // x,y,z,|x|^2 for tile cols j
  __shared__ int sBins[16 * 16];

  const int tid = threadIdx.x;
  const int i0 = blockIdx.y * 16;
  const int j0 = blockIdx.x * 16;

  // ---- stage weight.T (+ tile coords) into LDS --------------------------
  for (int idx = tid; idx < DIM * NBINS; idx += 256) {
    const int c = idx / NBINS;
    const int b = idx - c * NBINS;
    sWT[b * DIM + c] = weight[idx];  // coalesced global read
  }
  if (tid < 16) {
    const float xx = x[(i0 + tid) * 3 + 0];
    const float yy = x[(i0 + tid) * 3 + 1];
    const float zz = x[(i0 + tid) * 3 + 2];
    sXI[tid * 4 + 0] = xx;
    sXI[tid * 4 + 1] = yy;
    sXI[tid * 4 + 2] = zz;
    sXI[tid * 4 + 3] = xx * xx + yy * yy + zz * zz;
  } else if (tid < 32) {
    const int t = tid - 16;
    const float xx = x[(j0 + t) * 3 + 0];
    const float yy = x[(j0 + t) * 3 + 1];
    const float zz = x[(j0 + t) * 3 + 2];
    sXJ[t * 4 + 0] = xx;
    sXJ[t * 4 + 1] = yy;
    sXJ[t * 4 + 2] = zz;
    sXJ[t * 4 + 3] = xx * xx + yy * yy + zz * zz;
  }

  // per-lane channel constants (channels fixed per lane across all rows)
  const int lane = tid & 31;
  const int c = lane * 4;
  const float4 g4 = *(const float4*)(gamma_pair + c);
  const float4 bp4 = *(const float4*)(beta_pair + c);
  const float4 bi4 = *(const float4*)(bias + c);
  const float4 bb4 = make_float4(bp4.x + bi4.x, bp4.y + bi4.y,
                                 bp4.z + bi4.z, bp4.w + bi4.w);

  __syncthreads();

  // ---- step 1: wave 0 -> 16x16 dot tile via WMMA, then bins -------------
  if (tid < 32) {
    const int half = lane >> 4;   // 0: K=0,1   1: K=2,3
    const int mn = lane & 15;     // A: row M,  B: col N
    v2f a, b;
    if (half == 0) {
      a[0] = sXI[mn * 4 + 0];  // K=0 (x)
      a[1] = sXI[mn * 4 + 1];  // K=1 (y)
      b[0] = sXJ[mn * 4 + 0];
      b[1] = sXJ[mn * 4 + 1];
    } else {
      a[0] = sXI[mn * 4 + 2];  // K=2 (z)
      a[1] = 0.0f;             // K=3 (pad)
      b[0] = sXJ[mn * 4 + 2];
      b[1] = 0.0f;
    }

    float dotv[8];
#if defined(__has_builtin) && __has_builtin(__builtin_amdgcn_wmma_f32_16x16x4_f32)
    v8f acc = {};
    acc = __builtin_amdgcn_wmma_f32_16x16x4_f32(
        /*neg_a=*/false, a, /*neg_b=*/false, b,
        /*c_mod=*/(short)0, acc, /*reuse_a=*/false, /*reuse_b=*/false);
#pragma unroll
    for (int r = 0; r < 8; ++r) dotv[r] = acc[r];
#else
    // scalar fallback (keeps compiling; histogram will show wmma==0)
#pragma unroll
    for (int r = 0; r < 8; ++r) {
      const int M = r + (half << 3);
      dotv[r] = sXI[M * 4 + 0] * sXJ[mn * 4 + 0] +
                sXI[M * 4 + 1] * sXJ[mn * 4 + 1] +
                sXI[M * 4 + 2] * sXJ[mn * 4 + 2];
    }
#endif
    // D layout: VGPR r -> M=r (lanes 0-15), M=r+8 (lanes 16-31); N = lane&15
#pragma unroll
    for (int r = 0; r < 8; ++r) {
      const int M = r + (half << 3);
      float d2 = sXI[M * 4 + 3] + sXJ[mn * 4 + 3] - 2.0f * dotv[r];
      d2 = fmaxf(d2, 0.0f);
      const float dd = sqrtf(d2);
      // #edges (2.0 : 0.5 : 20.0) <= dd, clamped to 36
      int bin = (dd < 2.0f) ? 0 : ((int)floorf((dd - 2.0f) * 2.0f) + 1);
      bin = bin > 36 ? 36 : bin;
      sBins[M * 16 + mn] = bin;
    }
  }
  __syncthreads();

  // ---- step 2: stream the 256 pair rows of this tile --------------------
  const int wave = tid >> 5;
  const float* wrow_base = sWT;
#pragma unroll 1
  for (int r = wave; r < 256; r += 8) {
    const int li = r >> 4;
    const int lj = r & 15;
    const size_t rowOff =
        ((size_t)(i0 + li) * NSEQ + (size_t)(j0 + lj)) * DIM + c;

    float4 v = *(const float4*)(pair + rowOff);
    float s = v.x + v.y + v.z + v.w;
    float sq = v.x * v.x + v.y * v.y + v.z * v.z + v.w * v.w;
#pragma unroll
    for (int m = 16; m > 0; m >>= 1) {
      s += __shfl_xor(s, m, 32);
      sq += __shfl_xor(sq, m, 32);
    }
    const float mu = s * (1.0f / DIM);
    const float var = sq * (1.0f / DIM) - mu * mu;
    const float rstd = rsqrtf(var + LN_EPS);

    const int bin = sBins[r];
    const float4 e = *(const float4*)(wrow_base + bin * DIM + c);

    float4 o;
    o.x = (v.x - mu) * rstd * g4.x + bb4.x + e.x;
    o.y = (v.y - mu) * rstd * g4.y + bb4.y + e.y;
    o.z = (v.z - mu) * rstd * g4.z + bb4.z + e.z;
    o.w = (v.w - mu) * rstd * g4.w + bb4.w + e.w;
    *(float4*)(out_pair + rowOff) = o;
  }
}

// ---------------------------------------------------------------------------
extern "C" void kernel_launch(void* const* d_in, const int* in_sizes, int n_in,
                              void* d_out, int out_size, void* d_ws,
                              size_t ws_size, hipStream_t stream) {
  const float* x = (const float*)d_in[0];
  const float* pair = (const float*)d_in[1];
  const float* single = (const float*)d_in[2];
  const float* weight = (const float*)d_in[3];
  const float* bias = (const float*)d_in[4];
  const float* gamma_pair = (const float*)d_in[5];
  const float* beta_pair = (const float*)d_in[6];
  const float* gamma_msa = (const float*)d_in[7];
  const float* beta_msa = (const float*)d_in[8];

  float* out_single = (float*)d_out;                       // [1024,128]
  float* out_pair = out_single + (size_t)NSEQ * DIM;       // [1024,1024,128]

  ln_single_kernel<<<NSEQ / 8, 256, 0, stream>>>(single, gamma_msa, beta_msa,
                                                 out_single);
  dim3 grid(NSEQ / 16, NSEQ / 16);  // 64 x 64 tiles
  pair_kernel<<<grid, 256, 0, stream>>>(x, pair, weight, bias, gamma_pair,
                                        beta_pair, out_pair);
}